// GraphConv_14542759264284
// MI455X (gfx1250) — compile-verified
//
#include <hip/hip_runtime.h>
#include <stdint.h>

// ---------------------------------------------------------------------------
// GraphConv (LightGCN-style, 3 hops) for MI455X / gfx1250.
// Memory-bound SpMM-scatter (AI ~= 0.24 FLOP/B) -> optimize data movement:
//   * double-buffered async global->LDS staging of gathered rows (ASYNCcnt)
//   * 128-bit coalesced accesses, 16 lanes per edge (wave32: 2 edges/wave)
//   * native f32 global atomics for segment-sum
//   * non-temporal loads for streamed edge arrays
// WMMA is deliberately NOT used: there is no dense MAC structure here (the
// per-edge weight is a diagonal scale), so matrix pipes cannot beat the
// 23.3 TB/s HBM roofline that bounds this op.
// ---------------------------------------------------------------------------

#define N_USERS  100000
#define N_ITEMS  200000
#define N_NODES  300000
#define DIM      64
#define NNZ_E    1200000
#define NHOPS    3
#define EPB      16              // edges per LDS tile (256 thr = 16 thr/edge)
#define TILES    4               // tiles per block, pipelined 2-deep

__device__ __forceinline__ uint32_t hashu(uint32_t x) {
  x ^= x >> 16; x *= 0x7feb352dU;
  x ^= x >> 15; x *= 0x846ca68bU;
  x ^= x >> 16;
  return x;
}

// hop-0: concat(user,item) -> agg buffer and out[:, 0, :]
__global__ void k_init(const float* __restrict__ u, const float* __restrict__ it,
                       float* __restrict__ agg, float* __restrict__ out)
{
  size_t gi = (size_t)blockIdx.x * blockDim.x + threadIdx.x;   // float4 index
  size_t f  = gi * 4;
  size_t n  = f >> 6;
  int    dd = (int)(f & 63);
  float4 x;
  if (n < N_USERS) x = *reinterpret_cast<const float4*>(u + n * DIM + dd);
  else             x = *reinterpret_cast<const float4*>(it + (n - N_USERS) * (size_t)DIM + dd);
  *reinterpret_cast<float4*>(agg + f) = x;
  *reinterpret_cast<float4*>(out + n * (size_t)(DIM * (NHOPS + 1)) + dd) = x;
}

__global__ void k_zero(float* __restrict__ p)
{
  size_t gi = (size_t)blockIdx.x * blockDim.x + threadIdx.x;
  *reinterpret_cast<float4*>(p + gi * 4) = make_float4(0.f, 0.f, 0.f, 0.f);
}

// SpMM scatter: nxt[rows[e]] += v_e * cur[cols[e]], v_e = vals[e]*keep*2.
// 16 threads per edge own one 16B chunk of the 256B row. Gathered rows are
// staged through two LDS buffers with the gfx1250 async-copy path, pipelined
// so the HBM gather of tile i+1 overlaps the atomics of tile i.
__global__ void k_spmm(const float* __restrict__ cur, float* __restrict__ nxt,
                       const float* __restrict__ vals, const int* __restrict__ rows,
                       const int* __restrict__ cols, int hop)
{
  __shared__ float tile[2][EPB * DIM];              // 2 x 4 KB
  const int t    = threadIdx.x;                     // 0..255
  const int le   = t >> 4;                          // local edge 0..15
  const int ch   = t & 15;                          // float4 chunk 0..15
  const int base = blockIdx.x * (EPB * TILES) + le; // NNZ % (EPB*TILES) == 0

  // preload all edge data for this block's 4 tiles (one loadcnt drain total)
  int   c[TILES], r[TILES];
  float v[TILES];
#pragma unroll
  for (int i = 0; i < TILES; ++i) {
    const int e = base + i * EPB;
    c[i] = __builtin_nontemporal_load(cols + e);
    r[i] = __builtin_nontemporal_load(rows + e);
    v[i] = __builtin_nontemporal_load(vals + e);
    // edge dropout p=0.5, inverted-dropout rescale 1/(1-p) = 2
    const uint32_t h = hashu((uint32_t)e * 2654435761u +
                             (uint32_t)(hop + 1) * 0x85EBCA6Bu);
    v[i] = (h & 0x80000000u) ? (v[i] * 2.0f) : 0.0f;
  }

  const uint32_t lds0 = (uint32_t)(uintptr_t)(&tile[0][t * 4]);
  const uint32_t lds1 = (uint32_t)(uintptr_t)(&tile[1][t * 4]);

  // prologue: kick off tile 0's gather
  {
    const float* s0 = cur + (size_t)c[0] * DIM + ch * 4;
    asm volatile("global_load_async_to_lds_b128 %0, %1, off"
                 :: "v"(lds0), "v"(s0) : "memory");
  }

#pragma unroll
  for (int i = 0; i < TILES; ++i) {
    if (i + 1 < TILES) {
      // issue next tile's gather into the other buffer, then wait for tile i
      const float*   sn   = cur + (size_t)c[i + 1] * DIM + ch * 4;
      const uint32_t ldsn = ((i + 1) & 1) ? lds1 : lds0;
      asm volatile("global_load_async_to_lds_b128 %0, %1, off"
                   :: "v"(ldsn), "v"(sn) : "memory");
      asm volatile("s_wait_asynccnt 1" ::: "memory"); // in-order => tile i done
    } else {
      asm volatile("s_wait_asynccnt 0" ::: "memory");
    }
    const float* lp = (i & 1) ? &tile[1][t * 4] : &tile[0][t * 4];
    float4 x = *reinterpret_cast<const float4*>(lp);  // ds_load_b128
    float* dst = nxt + (size_t)r[i] * DIM + ch * 4;
    unsafeAtomicAdd(dst + 0, v[i] * x.x);
    unsafeAtomicAdd(dst + 1, v[i] * x.y);
    unsafeAtomicAdd(dst + 2, v[i] * x.z);
    unsafeAtomicAdd(dst + 3, v[i] * x.w);
  }
}

// message dropout (p=0.1, scale 1/0.9) fused with output-stack write
__global__ void k_drop(float* __restrict__ agg, float* __restrict__ out, int hop)
{
  size_t gi = (size_t)blockIdx.x * blockDim.x + threadIdx.x;
  size_t f  = gi * 4;
  float4 x  = *reinterpret_cast<float4*>(agg + f);
  float  r4[4] = {x.x, x.y, x.z, x.w};
  const uint32_t base = (uint32_t)f ^ ((uint32_t)(hop + 101) * 0x9E3779B9u);
#pragma unroll
  for (int i = 0; i < 4; ++i) {
    uint32_t h   = hashu(base + (uint32_t)i);
    float    u01 = (float)(h >> 8) * (1.0f / 16777216.0f);
    r4[i] = (u01 < 0.9f) ? r4[i] * (1.0f / 0.9f) : 0.0f;
  }
  x = make_float4(r4[0], r4[1], r4[2], r4[3]);
  *reinterpret_cast<float4*>(agg + f) = x;          // feeds next hop
  size_t n  = f >> 6;
  int    dd = (int)(f & 63);
  *reinterpret_cast<float4*>(out + n * (size_t)(DIM * (NHOPS + 1)) +
                             (size_t)(hop + 1) * DIM + dd) = x;
}

extern "C" void kernel_launch(void* const* d_in, const int* in_sizes, int n_in,
                              void* d_out, int out_size, void* d_ws, size_t ws_size,
                              hipStream_t stream)
{
  (void)in_sizes; (void)n_in; (void)out_size; (void)ws_size;
  const float* user = (const float*)d_in[0];
  const float* item = (const float*)d_in[1];
  const float* vals = (const float*)d_in[2];
  const int*   rows = (const int*)d_in[3];
  const int*   cols = (const int*)d_in[4];
  float*       out  = (float*)d_out;

  float* aggA = (float*)d_ws;                       // N*D f32
  float* aggB = aggA + (size_t)N_NODES * DIM;       // N*D f32 (ping-pong)

  const int ND4     = N_NODES * DIM / 4;            // 4,800,000 float4s
  const int EW_GRID = ND4 / 256;                    // 18,750 blocks
  const int SP_GRID = NNZ_E / (EPB * TILES);        // 18,750 blocks

  k_init<<<EW_GRID, 256, 0, stream>>>(user, item, aggA, out);

  float* cur = aggA;
  float* nxt = aggB;
  for (int hop = 0; hop < NHOPS; ++hop) {
    k_zero<<<EW_GRID, 256, 0, stream>>>(nxt);
    k_spmm<<<SP_GRID, 256, 0, stream>>>(cur, nxt, vals, rows, cols, hop);
    k_drop<<<EW_GRID, 256, 0, stream>>>(nxt, out, hop);
    float* tmp = cur; cur = nxt; nxt = tmp;
  }
}